// Net_66090956751513
// MI455X (gfx1250) — compile-verified
//
#include <hip/hip_runtime.h>
#include <hip/hip_bf16.h>
#include <math.h>

typedef float v2f __attribute__((ext_vector_type(2)));
typedef float v8f __attribute__((ext_vector_type(8)));

#define FIN  512
#define HID  64
#define NCLS 40

// ---------------- degree / normalization ----------------
__global__ void k_init_deg(float* __restrict__ deg, int n) {
  int i = blockIdx.x * blockDim.x + threadIdx.x;
  if (i < n) deg[i] = 1.0f;                       // self-loop contribution
}

__global__ void k_deg_scatter(const int* __restrict__ dst, float* __restrict__ deg, int e) {
  int i = blockIdx.x * blockDim.x + threadIdx.x;
  if (i < e) atomicAdd(&deg[dst[i]], 1.0f);
}

__global__ void k_dinv(float* __restrict__ deg, int n) {
  int i = blockIdx.x * blockDim.x + threadIdx.x;
  if (i < n) deg[i] = rsqrtf(deg[i]);             // deg >= 1 always (self loop)
}

__global__ void k_edge_norm(const int* __restrict__ src, const int* __restrict__ dst,
                            const float* __restrict__ dinv, float* __restrict__ nrm, int e) {
  int i = blockIdx.x * blockDim.x + threadIdx.x;
  if (i < e) nrm[i] = dinv[src[i]] * dinv[dst[i]];
}

// ---------------- GEMM1: H = X @ W1  (M=100000, K=512, N=64), fp32 WMMA ----------------
// 128 threads = 4 waves; each wave owns one 16-wide N tile; block owns a 16-row M tile.
__global__ void __launch_bounds__(128) k_gemm1_wmma(const float* __restrict__ X,
                                                    const float* __restrict__ W,
                                                    float* __restrict__ H) {
  __shared__ float lds[16 * 516];                 // stride 516: 516%64==4 -> conflict-free
  const int rowbase = blockIdx.x * 16;
  const int tid = threadIdx.x;

  // cooperative coalesced load of the 16x512 X tile (float4)
  for (int i = tid; i < 2048; i += 128) {
    int r  = i >> 7;                              // /128 float4s per row
    int c4 = i & 127;
    float4 v = *reinterpret_cast<const float4*>(X + (size_t)(rowbase + r) * FIN + (size_t)c4 * 4);
    float* p = &lds[r * 516 + c4 * 4];
    p[0] = v.x; p[1] = v.y; p[2] = v.z; p[3] = v.w;
  }
  __syncthreads();

  const int lane = tid & 31;
  const int half = lane >> 4;                     // 0: K lo pair, 1: K hi pair
  const int lm   = lane & 15;                     // M index (A) / N index (B,C)
  const int nb   = (tid >> 5) * 16;               // wave's N tile base

  v8f acc0 = {}, acc1 = {};
  for (int k0 = 0; k0 < FIN; k0 += 8) {
    // A fragment: lane lm holds row lm, K = k0+2*half, k0+2*half+1
    const float* ap = &lds[lm * 516 + k0 + 2 * half];
    v2f a0; a0.x = ap[0]; a0.y = ap[1];
    v2f a1; a1.x = ap[4]; a1.y = ap[5];
    // B fragment: lane lm holds col nb+lm, same K pair (W1 is hot in L2/L0)
    const int kk = k0 + 2 * half;
    v2f b0; b0.x = W[(size_t)kk       * HID + nb + lm];
            b0.y = W[(size_t)(kk + 1) * HID + nb + lm];
    v2f b1; b1.x = W[(size_t)(kk + 4) * HID + nb + lm];
            b1.y = W[(size_t)(kk + 5) * HID + nb + lm];
    acc0 = __builtin_amdgcn_wmma_f32_16x16x4_f32(false, a0, false, b0, (short)0, acc0, false, false);
    acc1 = __builtin_amdgcn_wmma_f32_16x16x4_f32(false, a1, false, b1, (short)0, acc1, false, false);
  }
  acc0 = acc0 + acc1;

  const int col = nb + lm;
#pragma unroll
  for (int r = 0; r < 8; ++r) {
    int row = rowbase + r + 8 * half;             // VGPR r: M=r (lanes 0-15) / M=r+8 (16-31)
    H[(size_t)row * HID + col] = acc0[r];
  }
}

// ---------------- aggregation layer 1 ----------------
__global__ void k_selfloop1(const float* __restrict__ H, const float* __restrict__ dinv,
                            float* __restrict__ A, int n) {
  int i = blockIdx.x * blockDim.x + threadIdx.x;  // i over n*HID
  if (i < n * HID) {
    int node = i >> 6;
    float d = dinv[node];
    A[i] = H[i] * d * d;
  }
}

__global__ void k_scatter1(const int* __restrict__ src, const int* __restrict__ dst,
                           const float* __restrict__ nrm, const float* __restrict__ H,
                           float* __restrict__ A, int e) {
  int i = blockIdx.x * blockDim.x + threadIdx.x;  // i over e*64
  if (i < e * HID) {
    int ei = i >> 6;
    int f  = i & 63;
    int s = src[ei], d = dst[ei];
    atomicAdd(&A[(size_t)d * HID + f], H[(size_t)s * HID + f] * nrm[ei]);
  }
}

__global__ void k_bias_relu(const float* __restrict__ A, const float* __restrict__ b1,
                            float* __restrict__ H2, int n) {
  int i = blockIdx.x * blockDim.x + threadIdx.x;
  if (i < n * HID) {
    float v = A[i] + b1[i & 63];
    H2[i] = v > 0.0f ? v : 0.0f;
  }
}

// ---------------- GEMM2: G = H2 @ W2  (M=100000, K=64, N=40 -> 3 tiles of 16) ----------------
__global__ void __launch_bounds__(96) k_gemm2_wmma(const float* __restrict__ H2,
                                                   const float* __restrict__ W,
                                                   float* __restrict__ G) {
  __shared__ float lds[16 * 68];                  // 68%64==4 -> conflict-free
  const int rowbase = blockIdx.x * 16;
  const int tid = threadIdx.x;

  for (int i = tid; i < 256; i += 96) {           // 16x64 tile as float4
    int r  = i >> 4;
    int c4 = i & 15;
    float4 v = *reinterpret_cast<const float4*>(H2 + (size_t)(rowbase + r) * HID + (size_t)c4 * 4);
    float* p = &lds[r * 68 + c4 * 4];
    p[0] = v.x; p[1] = v.y; p[2] = v.z; p[3] = v.w;
  }
  __syncthreads();

  const int lane = tid & 31;
  const int half = lane >> 4;
  const int lm   = lane & 15;
  const int n    = (tid >> 5) * 16 + lm;          // 0..47 (padded N)
  const int nc   = n < NCLS ? n : (NCLS - 1);     // clamp; mask keeps EXEC all-ones for WMMA
  const float msk = n < NCLS ? 1.0f : 0.0f;

  v8f acc = {};
  for (int k0 = 0; k0 < HID; k0 += 4) {
    const float* ap = &lds[lm * 68 + k0 + 2 * half];
    v2f a; a.x = ap[0]; a.y = ap[1];
    const int kk = k0 + 2 * half;
    v2f b; b.x = W[kk * NCLS + nc] * msk;
           b.y = W[(kk + 1) * NCLS + nc] * msk;
    acc = __builtin_amdgcn_wmma_f32_16x16x4_f32(false, a, false, b, (short)0, acc, false, false);
  }

  if (n < NCLS) {
#pragma unroll
    for (int r = 0; r < 8; ++r) {
      int row = rowbase + r + 8 * half;
      G[(size_t)row * NCLS + n] = acc[r];
    }
  }
}

// ---------------- aggregation layer 2 (into d_out) ----------------
__global__ void k_selfloop2(const float* __restrict__ G, const float* __restrict__ dinv,
                            float* __restrict__ out, int n) {
  int i = blockIdx.x * blockDim.x + threadIdx.x;  // i over n*40
  if (i < n * NCLS) {
    int node = i / NCLS;
    float d = dinv[node];
    out[i] = G[i] * d * d;
  }
}

__global__ void k_scatter2(const int* __restrict__ src, const int* __restrict__ dst,
                           const float* __restrict__ nrm, const float* __restrict__ G,
                           float* __restrict__ out, int e) {
  int i = blockIdx.x * blockDim.x + threadIdx.x;  // i over e*40
  if (i < e * NCLS) {
    int ei = i / NCLS;
    int f  = i - ei * NCLS;
    int s = src[ei], d = dst[ei];
    atomicAdd(&out[(size_t)d * NCLS + f], G[(size_t)s * NCLS + f] * nrm[ei]);
  }
}

// ---------------- in-place log_softmax over 40 classes, one wave32 per node ----------------
__global__ void __launch_bounds__(256) k_logsoftmax(float* __restrict__ out,
                                                    const float* __restrict__ b2, int n) {
  const int lane = threadIdx.x & 31;
  const int node = blockIdx.x * 8 + (threadIdx.x >> 5);
  if (node >= n) return;
  float* row = out + (size_t)node * NCLS;

  float v0 = row[lane] + b2[lane];                          // lane < 32 < 40 always valid
  float v1 = (lane < 8) ? (row[32 + lane] + b2[32 + lane]) : -1.0e30f;

  float m = fmaxf(v0, v1);
#pragma unroll
  for (int o = 16; o > 0; o >>= 1) m = fmaxf(m, __shfl_xor(m, o));

  float s = __expf(v0 - m) + ((lane < 8) ? __expf(v1 - m) : 0.0f);
#pragma unroll
  for (int o = 16; o > 0; o >>= 1) s += __shfl_xor(s, o);

  float lse = m + __logf(s);
  row[lane] = v0 - lse;
  if (lane < 8) row[32 + lane] = v1 - lse;
}

// ---------------- launcher ----------------
extern "C" void kernel_launch(void* const* d_in, const int* in_sizes, int n_in,
                              void* d_out, int out_size, void* d_ws, size_t ws_size,
                              hipStream_t stream) {
  (void)n_in; (void)out_size; (void)ws_size;
  const float* X  = (const float*)d_in[0];
  const int*   EI = (const int*)d_in[1];
  const float* W1 = (const float*)d_in[2];
  const float* b1 = (const float*)d_in[3];
  const float* W2 = (const float*)d_in[4];
  const float* b2 = (const float*)d_in[5];
  float* out = (float*)d_out;

  const int n = in_sizes[0] / FIN;        // 100000
  const int e = in_sizes[1] / 2;          // 1600000
  const int* src = EI;
  const int* dst = EI + e;

  // workspace layout (floats): dinv[n] | norm[e] | H[n*64] | A1[n*64] (A1 reused as G[n*40])
  float* wsf  = (float*)d_ws;
  float* dinv = wsf;
  float* nrm  = dinv + n;
  float* H    = nrm + e;
  float* A1   = H + (size_t)n * HID;
  float* G    = A1;                       // A1 dead after bias/relu; reuse for GEMM2 output

  const int B = 256;
  const int nb_n   = (n + B - 1) / B;
  const int nb_e   = (e + B - 1) / B;
  const int nb_nh  = (n * HID + B - 1) / B;
  const int nb_eh  = (int)(((long long)e * HID + B - 1) / B);
  const int nb_nc  = (n * NCLS + B - 1) / B;
  const int nb_ec  = (int)(((long long)e * NCLS + B - 1) / B);
  const int nb_m16 = n / 16;              // 6250, n is a multiple of 16
  const int nb_ls  = (n + 7) / 8;

  // degrees + symmetric normalization
  k_init_deg   <<<nb_n, B, 0, stream>>>(dinv, n);
  k_deg_scatter<<<nb_e, B, 0, stream>>>(dst, dinv, e);
  k_dinv       <<<nb_n, B, 0, stream>>>(dinv, n);
  k_edge_norm  <<<nb_e, B, 0, stream>>>(src, dst, dinv, nrm, e);

  // layer 1
  k_gemm1_wmma <<<nb_m16, 128, 0, stream>>>(X, W1, H);
  k_selfloop1  <<<nb_nh, B, 0, stream>>>(H, dinv, A1, n);
  k_scatter1   <<<nb_eh, B, 0, stream>>>(src, dst, nrm, H, A1, e);
  k_bias_relu  <<<nb_nh, B, 0, stream>>>(A1, b1, H, n);     // H2 overwrites H in place

  // layer 2
  k_gemm2_wmma <<<nb_m16, 96, 0, stream>>>(H, W2, G);
  k_selfloop2  <<<nb_nc, B, 0, stream>>>(G, dinv, out, n);
  k_scatter2   <<<nb_ec, B, 0, stream>>>(src, dst, nrm, G, out, e);
  k_logsoftmax <<<nb_ls, 256, 0, stream>>>(out, b2, n);
}